// Boundary_gt_67070209294986
// MI455X (gfx1250) — compile-verified
//
#include <hip/hip_runtime.h>
#include <hip/hip_bf16.h>
#include <stdint.h>

// Problem geometry (fixed by the reference).
#define H_DIM 1024
#define W_DIM 2048
#define TILE_W 64
#define TILE_H 32
#define PITCH  (TILE_W + 4)   // 68 ints per LDS halo row
#define ROWS   (TILE_H + 4)   // 36 LDS halo rows (2-pixel halo each side)
#define NTHREADS 256

typedef __attribute__((ext_vector_type(4))) unsigned int v4u;
typedef __attribute__((ext_vector_type(4))) int          v4i;
typedef __attribute__((ext_vector_type(8))) int          v8i;

__device__ __forceinline__ int imax(int a, int b) { return a > b ? a : b; }
__device__ __forceinline__ int imin(int a, int b) { return a < b ? a : b; }

#if __has_builtin(__builtin_amdgcn_tensor_load_to_lds)
#define HAVE_TDM 1
#else
#define HAVE_TDM 0
#endif

#if HAVE_TDM
// Build a 2D Tensor-DMA descriptor: compact LDS tile <-> strided global rect.
//   g0: count=1 | lds_addr | global_addr[56:0] | type=2 ("image")
//   g1: data_size=4B | tensor dims | tile dims | row stride (elements)
__device__ __forceinline__ void tdm_make_desc(v4u& g0, v8i& g1,
                                              unsigned long long ga,
                                              unsigned int lds_addr,
                                              int tile_d0, int tile_d1,
                                              int stride_elems)
{
    g0.x = 1u;                                         // count=1, user mode
    g0.y = lds_addr;                                   // LDS byte address
    g0.z = (unsigned int)(ga & 0xFFFFFFFFull);         // global_addr[31:0]
    g0.w = (unsigned int)((ga >> 32) & 0x01FFFFFFull)  // global_addr[56:32]
         | (2u << 30);                                 // type = "image"

    g1[0] = (int)(2u << 16);                           // data_size = 4 bytes
    g1[1] = (int)((W_DIM & 0xFFFF) << 16);             // tensor_dim0[15:0]
    g1[2] = (int)(((unsigned)W_DIM >> 16) & 0xFFFF)    // tensor_dim0[31:16]
          | (int)((H_DIM & 0xFFFF) << 16);             // tensor_dim1[15:0]
    g1[3] = (int)(((unsigned)H_DIM >> 16) & 0xFFFF)    // tensor_dim1[31:16]
          | (int)(tile_d0 << 16);                      // tile_dim0
    g1[4] = tile_d1;                                   // tile_dim1 (tile_dim2 = 0)
    g1[5] = stride_elems;                              // tensor_dim0_stride[31:0]
    g1[6] = 0;                                         // stride hi / dim1_stride lo
    g1[7] = 0;
}
#endif

__global__ __launch_bounds__(NTHREADS)
void boundary_edge_kernel(const int* __restrict__ lab, float* __restrict__ out)
{
    __shared__ int   tile[ROWS * PITCH];        // label halo tile (9792 B)
    __shared__ float otile[TILE_H * TILE_W];    // edge output tile (8192 B)

    const int tid = threadIdx.x;
    const int x0 = blockIdx.x * TILE_W;         // output tile origin (image coords)
    const int y0 = blockIdx.y * TILE_H;
    const size_t img = (size_t)blockIdx.z * (size_t)H_DIM * (size_t)W_DIM;

    const bool interior = (x0 >= 2) && (x0 + TILE_W + 2 <= W_DIM) &&
                          (y0 >= 2) && (y0 + TILE_H + 2 <= H_DIM);

#if HAVE_TDM
    if (interior) {
        // One TDM DMA per workgroup: 68x36 int32 halo tile, global -> LDS.
        if (tid < 32) {  // wave 0 only; TDM issues once per wave (EXEC ignored)
            const unsigned long long ga =
                (unsigned long long)(uintptr_t)(lab + img +
                    (size_t)(y0 - 2) * W_DIM + (size_t)(x0 - 2));
            v4u g0; v8i g1;
            tdm_make_desc(g0, g1, ga, (unsigned int)(uintptr_t)&tile[0],
                          PITCH, ROWS, W_DIM);
            v4i gz = {0, 0, 0, 0};               // groups 2/3 unused (2D tensor)
#if defined(__clang_major__) && (__clang_major__ >= 23)
            v8i gz8 = {0, 0, 0, 0, 0, 0, 0, 0};
            __builtin_amdgcn_tensor_load_to_lds(g0, g1, gz, gz, gz8, 0);
#else
            __builtin_amdgcn_tensor_load_to_lds(g0, g1, gz, gz, 0);
#endif
            __builtin_amdgcn_s_wait_tensorcnt((short)0);
        }
    } else
#endif
    {
        // Border tiles (and fallback): bounds-checked cooperative fill.
        // Sentinel -1: identity for max (labels >= 0); (v & 127) -> 127 is
        // identity for min (labels <= 19).
        for (int idx = tid; idx < ROWS * PITCH; idx += NTHREADS) {
            const int ly = idx / PITCH, lx = idx % PITCH;
            const int gy = y0 - 2 + ly, gx = x0 - 2 + lx;
            int v = -1;
            if (gy >= 0 && gy < H_DIM && gx >= 0 && gx < W_DIM)
                v = lab[img + (size_t)gy * W_DIM + gx];
            tile[idx] = v;
        }
    }
    __syncthreads();

    // 13-point Euclidean-disk (r=2) min/max stencil from LDS.
    const int lx  = tid & (TILE_W - 1);   // 0..63
    const int ty0 = tid >> 6;             // 0..3
#pragma unroll
    for (int r = 0; r < TILE_H / 4; ++r) {
        const int ly = ty0 + r * 4;
        const int gy = y0 + ly;
        const int gx = x0 + lx;
        const int c  = (ly + 2) * PITCH + (lx + 2);

        int a = tile[c];
        int M = a, m = a & 127;
#define ACC(off) { int a_ = tile[c + (off)]; M = imax(M, a_); m = imin(m, a_ & 127); }
        ACC(-2)  ACC(-1)  ACC(+1)  ACC(+2)            // dy = 0
        ACC(-PITCH - 1)  ACC(-PITCH)  ACC(-PITCH + 1) // dy = -1
        ACC(+PITCH - 1)  ACC(+PITCH)  ACC(+PITCH + 1) // dy = +1
        ACC(-2 * PITCH)  ACC(+2 * PITCH)              // dy = +/-2
#undef ACC

        // Disk reaches the zero-pad ring iff within 2 of the image border.
        const bool nearb = (gy < 2) | (gy >= H_DIM - 2) | (gx < 2) | (gx >= W_DIM - 2);
        const bool edge  = (M > 0) && ((m != M) || nearb);
#if HAVE_TDM
        otile[ly * TILE_W + lx] = edge ? 1.0f : 0.0f;
#else
        out[img + (size_t)gy * W_DIM + gx] = edge ? 1.0f : 0.0f;
#endif
    }

#if HAVE_TDM
    __syncthreads();
    // Output tile is always fully in-bounds and compact: one TDM store
    // per workgroup, LDS -> strided global. S_ENDPGM's implicit wait-idle
    // (all counters, incl. TENSORcnt) keeps LDS alive until the DMA lands.
    if (tid < 32) {
        const unsigned long long ga =
            (unsigned long long)(uintptr_t)(out + img +
                (size_t)y0 * W_DIM + (size_t)x0);
        v4u g0; v8i g1;
        tdm_make_desc(g0, g1, ga, (unsigned int)(uintptr_t)&otile[0],
                      TILE_W, TILE_H, W_DIM);
        v4i gz = {0, 0, 0, 0};
#if defined(__clang_major__) && (__clang_major__ >= 23)
        v8i gz8 = {0, 0, 0, 0, 0, 0, 0, 0};
        __builtin_amdgcn_tensor_store_from_lds(g0, g1, gz, gz, gz8, 0);
#else
        __builtin_amdgcn_tensor_store_from_lds(g0, g1, gz, gz, 0);
#endif
    }
#endif
}

extern "C" void kernel_launch(void* const* d_in, const int* in_sizes, int n_in,
                              void* d_out, int out_size, void* d_ws, size_t ws_size,
                              hipStream_t stream)
{
    const int* lab = (const int*)d_in[0];
    float* out = (float*)d_out;
    int B = in_sizes[0] / (H_DIM * W_DIM);
    if (B < 1) B = 1;
    dim3 grid(W_DIM / TILE_W, H_DIM / TILE_H, B);
    boundary_edge_kernel<<<grid, NTHREADS, 0, stream>>>(lab, out);
}